// RotatedARSLLoss_82471962018520
// MI455X (gfx1250) — compile-verified
//
#include <hip/hip_runtime.h>
#include <math.h>

#define NTOT 87296
#define NC 16
#define KPOS 256
#define KHN 2048
#define NBLK ((NTOT + 255) / 256)   // 341, exact: 341*256 == 87296
#define PI_F 3.14159265358979323846f
#define HPI_F 1.57079632679489661923f
#define EPSF 1e-8f

typedef float v2f __attribute__((ext_vector_type(2)));
typedef float v8f __attribute__((ext_vector_type(8)));

struct Feat {
  const float* cls[5];
  const float* box[5];
  const float* ang[5];
  const float* ctr[5];
};

// ---------------------------------------------------------------- helpers
__device__ __forceinline__ float sigmoidf_(float x) { return 1.f / (1.f + expf(-x)); }

__device__ __forceinline__ unsigned fkey(float f) {
  unsigned u = __float_as_uint(f);
  return (u & 0x80000000u) ? ~u : (u | 0x80000000u);
}

__device__ __forceinline__ void lvl_of(int n, int& lvl, int& off, int& w, float& stride) {
  if (n < 65536)      { lvl = 0; off = n;         w = 256; stride = 8.f;   }
  else if (n < 81920) { lvl = 1; off = n - 65536; w = 128; stride = 16.f;  }
  else if (n < 86016) { lvl = 2; off = n - 81920; w = 64;  stride = 32.f;  }
  else if (n < 87040) { lvl = 3; off = n - 86016; w = 32;  stride = 64.f;  }
  else                { lvl = 4; off = n - 87040; w = 16;  stride = 128.f; }
}

__device__ __forceinline__ void point_of(int n, float& px, float& py, int& lvl, int& off, int& w) {
  float s; lvl_of(n, lvl, off, w, s);
  int x = off % w, y = off / w;
  px = ((float)x + 0.5f) * s;
  py = ((float)y + 0.5f) * s;
}

__device__ __forceinline__ void decode5(float px, float py, const float* p, float* o) {
  float l = p[0], t = p[1], r = p[2], b = p[3], a = p[4];
  float ca = cosf(a), sa = sinf(a);
  float w = l + r, h = t + b;
  float oxl = (r - l) * 0.5f, oyl = (b - t) * 0.5f;
  float ox = ca * oxl - sa * oyl;
  float oy = sa * oxl + ca * oyl;
  float an = fmodf(a + HPI_F, PI_F);
  if (an < 0.f) an += PI_F;
  an -= HPI_F;
  o[0] = px + ox; o[1] = py + oy; o[2] = w; o[3] = h; o[4] = an;
}

__device__ __forceinline__ void encode5(float px, float py, const float* g, float* o) {
  float cx = g[0], cy = g[1], w = g[2], h = g[3], a = g[4];
  float ca = cosf(a), sa = sinf(a);
  float dx = px - cx, dy = py - cy;
  float ox = ca * dx + sa * dy;
  float oy = -sa * dx + ca * dy;
  o[0] = w * 0.5f + ox; o[1] = h * 0.5f + oy;
  o[2] = w * 0.5f - ox; o[3] = h * 0.5f - oy; o[4] = a;
}

__device__ __forceinline__ bool inside_box(float px, float py, const float* b) {
  float ca = cosf(b[4]), sa = sinf(b[4]);
  float dx = px - b[0], dy = py - b[1];
  float ox = ca * dx + sa * dy;
  float oy = -sa * dx + ca * dy;
  float m = fminf(fminf(b[2] * 0.5f + ox, b[2] * 0.5f - ox),
                  fminf(b[3] * 0.5f + oy, b[3] * 0.5f - oy));
  return m > 0.f;
}

__device__ __forceinline__ void corners4(const float* b, float* c) {
  const float fx[4] = {-0.5f, 0.5f, 0.5f, -0.5f};
  const float fy[4] = {-0.5f, -0.5f, 0.5f, 0.5f};
  float ca = cosf(b[4]), sa = sinf(b[4]);
  for (int k = 0; k < 4; k++) {
    float dx = fx[k] * b[2], dy = fy[k] * b[3];
    c[2 * k]     = b[0] + dx * ca - dy * sa;
    c[2 * k + 1] = b[1] + dx * sa + dy * ca;
  }
}

__device__ float quad_inter_area(const float* A, const float* B) {
  float dax[4], day[4], dbx[4], dby[4];
  for (int i = 0; i < 4; i++) {
    int i1 = (i + 1) & 3;
    dax[i] = A[2 * i1] - A[2 * i];   day[i] = A[2 * i1 + 1] - A[2 * i + 1];
    dbx[i] = B[2 * i1] - B[2 * i];   dby[i] = B[2 * i1 + 1] - B[2 * i + 1];
  }
  float px[24], py[24]; bool val[24];
  for (int i = 0; i < 4; i++)
    for (int j = 0; j < 4; j++) {
      float dx = B[2 * j] - A[2 * i], dy = B[2 * j + 1] - A[2 * i + 1];
      float den = dax[i] * dby[j] - day[i] * dbx[j];
      float ds = (fabsf(den) < 1e-10f) ? 1.f : den;
      float t = (dx * dby[j] - dy * dbx[j]) / ds;
      float u = (dx * day[i] - dy * dax[i]) / ds;
      bool ok = (fabsf(den) > 1e-10f) && (t >= 0.f) && (t <= 1.f) && (u >= 0.f) && (u <= 1.f);
      int id = i * 4 + j;
      px[id] = A[2 * i] + t * dax[i];
      py[id] = A[2 * i + 1] + t * day[i];
      val[id] = ok;
    }
  for (int i = 0; i < 4; i++) {       // corners of A inside B
    bool allp = true, alln = true;
    for (int j = 0; j < 4; j++) {
      float s = dbx[j] * (A[2 * i + 1] - B[2 * j + 1]) - dby[j] * (A[2 * i] - B[2 * j]);
      allp = allp && (s >= -1e-9f); alln = alln && (s <= 1e-9f);
    }
    px[16 + i] = A[2 * i]; py[16 + i] = A[2 * i + 1]; val[16 + i] = allp || alln;
  }
  for (int i = 0; i < 4; i++) {       // corners of B inside A
    bool allp = true, alln = true;
    for (int j = 0; j < 4; j++) {
      float s = dax[j] * (B[2 * i + 1] - A[2 * j + 1]) - day[j] * (B[2 * i] - A[2 * j]);
      allp = allp && (s >= -1e-9f); alln = alln && (s <= 1e-9f);
    }
    px[20 + i] = B[2 * i]; py[20 + i] = B[2 * i + 1]; val[20 + i] = allp || alln;
  }
  int n = 0; float cx = 0.f, cy = 0.f;
  for (int k = 0; k < 24; k++) if (val[k]) { n++; cx += px[k]; cy += py[k]; }
  float inv = 1.f / (float)(n < 1 ? 1 : n);
  cx *= inv; cy *= inv;
  for (int k = 0; k < 24; k++) if (!val[k]) { px[k] = cx; py[k] = cy; }
  float ang[24]; int ord[24];
  for (int k = 0; k < 24; k++) { ang[k] = atan2f(py[k] - cy, px[k] - cx); ord[k] = k; }
  for (int i = 1; i < 24; i++) {      // stable insertion sort (argsort is stable)
    int o = ord[i]; float a = ang[o]; int j = i - 1;
    while (j >= 0 && ang[ord[j]] > a) { ord[j + 1] = ord[j]; j--; }
    ord[j + 1] = o;
  }
  float s = 0.f;
  for (int i = 0; i < 24; i++) {
    int a0 = ord[i], a1 = ord[(i + 1) % 24];
    float ax = px[a0] - cx, ay = py[a0] - cy;
    float bx = px[a1] - cx, by = py[a1] - cy;
    s += ax * by - ay * bx;
  }
  return (n >= 3) ? 0.5f * fabsf(s) : 0.f;
}

__device__ __forceinline__ float bce_(float p, float t) {
  float pc = fminf(fmaxf(p, 1e-12f), 1.f - 1e-12f);
  return -(t * logf(pc) + (1.f - t) * log1pf(-pc));
}

// ---------------------------------------------------------------- kernels
__global__ void k_convert(Feat tf, float* sig_cls, float* tbox, float* sig_iou,
                          float* maxv, float* rowmax, int* clsind,
                          float* p_cnt, float* p_sum, float* p_sq) {
  int n = blockIdx.x * 256 + threadIdx.x;
  float mv = -1.f;
  if (n < NTOT) {
    int lvl, off, w; float st;
    lvl_of(n, lvl, off, w, st);
    int hw = w * w;
    float si = sigmoidf_(tf.ctr[lvl][off]);
    sig_iou[n] = si;
    float best = -1.f; int bi = 0;
    for (int c = 0; c < NC; c++) {
      float v = sigmoidf_(tf.cls[lvl][c * hw + off]);
      sig_cls[n * NC + c] = v;
      if (v > best) { best = v; bi = c; }
    }
    rowmax[n] = best; clsind[n] = bi;
    mv = si * best;
    maxv[n] = mv;
    for (int j = 0; j < 4; j++) tbox[n * 5 + j] = tf.box[lvl][j * hw + off];
    tbox[n * 5 + 4] = tf.ang[lvl][off];
  }
  __shared__ float rc[256], rs[256], rq[256];
  int t = threadIdx.x;
  bool cand = (n < NTOT) && (mv >= 0.1f);
  rc[t] = cand ? 1.f : 0.f;
  rs[t] = cand ? mv : 0.f;
  rq[t] = cand ? mv * mv : 0.f;
  __syncthreads();
  for (int s = 128; s > 0; s >>= 1) {
    if (t < s) { rc[t] += rc[t + s]; rs[t] += rs[t + s]; rq[t] += rq[t + s]; }
    __syncthreads();
  }
  if (t == 0) { p_cnt[blockIdx.x] = rc[0]; p_sum[blockIdx.x] = rs[0]; p_sq[blockIdx.x] = rq[0]; }
}

__global__ void k_stats(const float* p_cnt, const float* p_sum, const float* p_sq, float* sc) {
  __shared__ float rc[512], rs[512], rq[512];
  int t = threadIdx.x;
  float a = 0.f, b = 0.f, c = 0.f;
  for (int i = t; i < NBLK; i += 512) { a += p_cnt[i]; b += p_sum[i]; c += p_sq[i]; }
  rc[t] = a; rs[t] = b; rq[t] = c; __syncthreads();
  for (int s = 256; s > 0; s >>= 1) {
    if (t < s) { rc[t] += rc[t + s]; rs[t] += rs[t + s]; rq[t] += rq[t + s]; }
    __syncthreads();
  }
  if (t == 0) {
    float n = rc[0], sum = rs[0], sq = rq[0];
    float mean = sum / fmaxf(n, 1.f);
    float var = (sq - 2.f * mean * sum + n * mean * mean) / fmaxf(n - 1.f, 1.f);
    float th = (n > 0.f) ? fminf(mean + sqrtf(fmaxf(var, 0.f)), 0.4f) : __builtin_inff();
    sc[0] = th; sc[1] = n; sc[2] = mean; sc[3] = var;
  }
}

__global__ void k_masks(const float* maxv, const float* sc, unsigned char* pos,
                        unsigned char* hn, float* p_pos) {
  int n = blockIdx.x * 256 + threadIdx.x;
  float th = sc[0];
  bool p = false;
  if (n < NTOT) {
    float mv = maxv[n];
    bool cand = mv >= 0.1f;
    p = mv >= th;
    pos[n] = p ? 1 : 0;
    hn[n] = (cand && (mv < th)) ? 1 : 0;
  }
  __shared__ float rp[256];
  int t = threadIdx.x;
  rp[t] = p ? 1.f : 0.f; __syncthreads();
  for (int s = 128; s > 0; s >>= 1) { if (t < s) rp[t] += rp[t + s]; __syncthreads(); }
  if (t == 0) p_pos[blockIdx.x] = rp[0];
}

__global__ void k_fallback(const float* p_pos, float* sc, const float* maxv,
                           unsigned char* pos, unsigned long long* keys) {
  __shared__ float rp[1024];
  __shared__ unsigned long long sk[1024];
  __shared__ int anyflag;
  int t = threadIdx.x;
  float a = 0.f;
  for (int i = t; i < NBLK; i += 1024) a += p_pos[i];
  rp[t] = a; __syncthreads();
  for (int s = 512; s > 0; s >>= 1) { if (t < s) rp[t] += rp[t + s]; __syncthreads(); }
  if (t == 0) { sc[4] = rp[0]; anyflag = (rp[0] > 0.f) ? 1 : 0; }
  __syncthreads();
  if (anyflag) return;
  for (int n = t; n < NTOT; n += 1024)
    keys[n] = (((unsigned long long)fkey(maxv[n])) << 32) | (unsigned long long)(0xFFFFFFFFu - (unsigned)n);
  __syncthreads();
  for (int it = 0; it < 10; it++) {
    unsigned long long best = 0ULL;
    for (int n = t; n < NTOT; n += 1024) { unsigned long long v = keys[n]; if (v > best) best = v; }
    sk[t] = best; __syncthreads();
    for (int s = 512; s > 0; s >>= 1) { if (t < s) { if (sk[t + s] > sk[t]) sk[t] = sk[t + s]; } __syncthreads(); }
    if (t == 0) {
      int idx = (int)(0xFFFFFFFFu - (unsigned)(sk[0] & 0xFFFFFFFFull));
      pos[idx] = 1; keys[idx] = 0ULL;
    }
    __syncthreads();
  }
}

__global__ void k_targets(const unsigned char* pos, const float* tbox, float* loct, float* locm) {
  int n = blockIdx.x * 256 + threadIdx.x;
  if (n >= NTOT) return;
  bool p = pos[n];
  for (int j = 0; j < 5; j++) loct[n * 5 + j] = p ? tbox[n * 5 + j] : 0.f;
  locm[n] = p ? 1.f : 0.f;
}

__global__ void k_mkkeys(const float* maxv, const unsigned char* mask, unsigned long long* keys) {
  int n = blockIdx.x * 256 + threadIdx.x;
  if (n >= NTOT) return;
  float v = mask[n] ? maxv[n] : -1.f;
  keys[n] = (((unsigned long long)fkey(v)) << 32) | (unsigned long long)(0xFFFFFFFFu - (unsigned)n);
}

__global__ void k_select(unsigned long long* keys, int K, int* oidx) {
  __shared__ unsigned long long sk[1024];
  int t = threadIdx.x;
  for (int k = 0; k < K; k++) {
    unsigned long long best = 0ULL;
    for (int n = t; n < NTOT; n += 1024) { unsigned long long v = keys[n]; if (v > best) best = v; }
    sk[t] = best; __syncthreads();
    for (int s = 512; s > 0; s >>= 1) { if (t < s) { if (sk[t + s] > sk[t]) sk[t] = sk[t + s]; } __syncthreads(); }
    if (t == 0) {
      int idx = (int)(0xFFFFFFFFu - (unsigned)(sk[0] & 0xFFFFFFFFull));
      oidx[k] = idx; keys[idx] = 0ULL;
    }
    __syncthreads();
  }
}

__global__ void k_gpos(const int* pidx, const unsigned char* pos, const float* maxv,
                       const int* clsind, const float* tbox,
                       int* pvalid, float* pv, float* pbox, int* pcls, float* pscl, float* Bpad) {
  int k = threadIdx.x;
  int i = pidx[k];
  bool valid = pos[i];
  pvalid[k] = valid ? 1 : 0;
  pv[k] = valid ? maxv[i] : -1.f;
  pcls[k] = clsind[i];
  float px, py; int lvl, off, w;
  point_of(i, px, py, lvl, off, w);
  pscl[k] = (float)lvl;
  float dec[5];
  decode5(px, py, &tbox[i * 5], dec);
  for (int j = 0; j < 5; j++) { pbox[k * 5 + j] = dec[j]; Bpad[k * 16 + j] = dec[j]; }
  Bpad[k * 16 + 5] = 1.f;
  for (int j = 6; j < 16; j++) Bpad[k * 16 + j] = 0.f;
}

__global__ void k_ghn(const int* hidx, const unsigned char* hn, const float* tbox, const int* clsind,
                      int* hvalid, float* hpts, float* hbox, int* hcls, float* hscl) {
  int h = blockIdx.x * 256 + threadIdx.x;
  if (h >= KHN) return;
  int i = hidx[h];
  hvalid[h] = hn[i] ? 1 : 0;
  hcls[h] = clsind[i];
  float px, py; int lvl, off, w;
  point_of(i, px, py, lvl, off, w);
  hscl[h] = (float)lvl;
  hpts[h * 2] = px; hpts[h * 2 + 1] = py;
  float dec[5];
  decode5(px, py, &tbox[i * 5], dec);
  for (int j = 0; j < 5; j++) hbox[h * 5 + j] = dec[j];
}

__global__ void k_iou(const float* hbox, const float* hpts, const int* hvalid, const int* hcls, const float* hscl,
                      const float* pbox, const int* pvalid, const float* pv, const int* pcls, const float* pscl,
                      float* W, int* potential) {
  int h = blockIdx.x;
  int k = threadIdx.x;
  float hb[5], pb[5];
  for (int j = 0; j < 5; j++) { hb[j] = hbox[h * 5 + j]; pb[j] = pbox[k * 5 + j]; }
  float cA[8], cB[8];
  corners4(hb, cA);
  corners4(pb, cB);
  float inter = quad_inter_area(cA, cB);
  float a1 = fabsf(hb[2] * hb[3]);
  float a2 = fabsf(pb[2] * pb[3]);
  float iou = inter / (a1 + a2 - inter + EPSF);
  float hx = hpts[h * 2], hy = hpts[h * 2 + 1];
  bool valid = (iou >= 0.6f) && (hcls[h] == pcls[k]) && inside_box(hx, hy, pb)
               && (fabsf(hscl[h] - pscl[k]) <= 1.f) && hvalid[h] && pvalid[k];
  W[h * KPOS + k] = valid ? pv[k] : 0.f;
  __shared__ float sm[256];
  sm[k] = valid ? iou : 0.f;
  __syncthreads();
  for (int s = 128; s > 0; s >>= 1) { if (k < s) sm[k] = fmaxf(sm[k], sm[k + s]); __syncthreads(); }
  if (k == 0) potential[h] = (sm[0] > 0.f) ? 1 : 0;
}

// agg = W[2048x256] x Bpad[256x16]  via chained V_WMMA_F32_16X16X4_F32
// A 16x4 f32 layout: lanes 0-15 -> M=0..15; VGPR0 holds K=(k0 / k0+2), VGPR1 K=(k0+1 / k0+3)
// D 16x16 f32 layout: VGPR v -> M=v (lanes 0-15), M=v+8 (lanes 16-31), N = lane&15
__global__ void k_wmma(const float* __restrict__ W, const float* __restrict__ B, float* __restrict__ aggr) {
  int lane = threadIdx.x;
  int m = lane & 15;
  int hi = lane >> 4;
  int row = blockIdx.x * 16 + m;
  v8f c = {0.f, 0.f, 0.f, 0.f, 0.f, 0.f, 0.f, 0.f};
  for (int k0 = 0; k0 < KPOS; k0 += 4) {
    int ka = k0 + 2 * hi;
    v2f a, b;
    a.x = W[row * KPOS + ka];
    a.y = W[row * KPOS + ka + 1];
    b.x = B[ka * 16 + m];
    b.y = B[(ka + 1) * 16 + m];
    c = __builtin_amdgcn_wmma_f32_16x16x4_f32(false, a, false, b, (short)0, c, false, false);
  }
  int rb = blockIdx.x * 16;
  for (int v = 0; v < 8; v++)
    aggr[(rb + v + 8 * hi) * 16 + m] = c[v];
}

__global__ void k_aggfin(const int* potential, const float* aggr, const float* hpts,
                         const int* hidx, float* loct, float* locm) {
  int h = blockIdx.x * 256 + threadIdx.x;
  if (h >= KHN) return;
  if (!potential[h]) return;
  float d = fmaxf(aggr[h * 16 + 5], EPSF);
  float bb[5];
  for (int j = 0; j < 5; j++) bb[j] = aggr[h * 16 + j] / d;
  float enc[5];
  encode5(hpts[h * 2], hpts[h * 2 + 1], bb, enc);
  int i = hidx[h];   // selection indices are distinct -> no write conflicts
  for (int j = 0; j < 5; j++) loct[i * 5 + j] = enc[j];
  locm[i] = 1.f;
}

__global__ void k_fsum(const unsigned char* pos, const unsigned char* hn, const float* rowmax,
                       const float* locm, float* locw, float* pc, float* pl, float* pi_) {
  int n = blockIdx.x * 256 + threadIdx.x;
  float a = 0.f, b = 0.f, c = 0.f;
  if (n < NTOT) {
    bool keep = pos[n] || hn[n];
    float ctm = keep ? rowmax[n] : 0.f;
    bool lp = locm[n] > 0.f;
    float lw = lp ? ctm : 0.f;
    locw[n] = lw;
    a = pos[n] ? ctm : 0.f;
    b = lw;
    c = lp ? 1.f : 0.f;
  }
  __shared__ float ra[256], rb[256], rc[256];
  int t = threadIdx.x;
  ra[t] = a; rb[t] = b; rc[t] = c; __syncthreads();
  for (int s = 128; s > 0; s >>= 1) {
    if (t < s) { ra[t] += ra[t + s]; rb[t] += rb[t + s]; rc[t] += rc[t + s]; }
    __syncthreads();
  }
  if (t == 0) { pc[blockIdx.x] = ra[0]; pl[blockIdx.x] = rb[0]; pi_[blockIdx.x] = rc[0]; }
}

__global__ void k_fsfin(const float* pc, const float* pl, const float* pi_, float* sc) {
  __shared__ float ra[512], rb[512], rc[512];
  int t = threadIdx.x;
  float a = 0.f, b = 0.f, c = 0.f;
  for (int i = t; i < NBLK; i += 512) { a += pc[i]; b += pl[i]; c += pi_[i]; }
  ra[t] = a; rb[t] = b; rc[t] = c; __syncthreads();
  for (int s = 256; s > 0; s >>= 1) {
    if (t < s) { ra[t] += ra[t + s]; rb[t] += rb[t + s]; rc[t] += rc[t + s]; }
    __syncthreads();
  }
  if (t == 0) { sc[5] = ra[0]; sc[6] = rb[0]; sc[7] = rc[0]; }
}

__global__ void k_student(Feat sf, const unsigned char* pos, const unsigned char* hn,
                          const float* sig_cls, const float* sig_iou_t,
                          const float* loct, const float* locm, const float* locw,
                          float* lc, float* lb, float* li) {
  int n = blockIdx.x * 256 + threadIdx.x;
  float lcn = 0.f, lbn = 0.f, lin = 0.f;
  if (n < NTOT) {
    int lvl, off, w; float st;
    lvl_of(n, lvl, off, w, st);
    int hw = w * w;
    float s_iou = sigmoidf_(sf.ctr[lvl][off]);
    bool keep = pos[n] || hn[n];
    for (int c = 0; c < NC; c++) {
      float jc = sigmoidf_(sf.cls[lvl][c * hw + off]) * s_iou;
      float tt = keep ? sig_cls[n * NC + c] : 0.f;
      float e = fabsf(jc - tt);
      lcn += bce_(jc, tt) * e * e;
    }
    if (locm[n] > 0.f) lin = bce_(s_iou, sig_iou_t[n]);
    float lw = locw[n];
    if (lw > 0.f) {
      float acc = 0.f;
      for (int j = 0; j < 5; j++) {
        float sv = (j < 4) ? sf.box[lvl][j * hw + off] : sf.ang[lvl][off];
        float d = fabsf(sv - loct[n * 5 + j]);
        acc += (d < 1.f) ? 0.5f * d * d : d - 0.5f;
      }
      lbn = lw * acc;
    }
  }
  __shared__ float ra[256], rb[256], rc[256];
  int t = threadIdx.x;
  ra[t] = lcn; rb[t] = lbn; rc[t] = lin; __syncthreads();
  for (int s = 128; s > 0; s >>= 1) {
    if (t < s) { ra[t] += ra[t + s]; rb[t] += rb[t + s]; rc[t] += rc[t + s]; }
    __syncthreads();
  }
  if (t == 0) { lc[blockIdx.x] = ra[0]; lb[blockIdx.x] = rb[0]; li[blockIdx.x] = rc[0]; }
}

__global__ void k_out(const float* lc, const float* lb, const float* li, const float* sc, float* out) {
  __shared__ float ra[512], rb[512], rc[512];
  int t = threadIdx.x;
  float a = 0.f, b = 0.f, c = 0.f;
  for (int i = t; i < NBLK; i += 512) { a += lc[i]; b += lb[i]; c += li[i]; }
  ra[t] = a; rb[t] = b; rc[t] = c; __syncthreads();
  for (int s = 256; s > 0; s >>= 1) {
    if (t < s) { ra[t] += ra[t + s]; rb[t] += rb[t + s]; rc[t] += rc[t + s]; }
    __syncthreads();
  }
  if (t == 0) {
    out[0] = ra[0] / fmaxf(sc[5], EPSF);   // loss_cls / max(cls_avg, eps)
    out[1] = rb[0] / fmaxf(sc[6], EPSF);   // loss_box / max(loc_avg, eps)
    out[2] = rc[0] / fmaxf(sc[7], 1.f);    // loss_iou / max(iou_avg, 1)
  }
}

// ---------------------------------------------------------------- host
extern "C" void kernel_launch(void* const* d_in, const int* in_sizes, int n_in,
                              void* d_out, int out_size, void* d_ws, size_t ws_size,
                              hipStream_t stream) {
  (void)in_sizes; (void)n_in; (void)out_size; (void)ws_size;
  Feat tf, sf;
  for (int i = 0; i < 5; i++) {
    tf.cls[i] = (const float*)d_in[0 + 4 * i];
    tf.box[i] = (const float*)d_in[1 + 4 * i];
    tf.ang[i] = (const float*)d_in[2 + 4 * i];
    tf.ctr[i] = (const float*)d_in[3 + 4 * i];
    sf.cls[i] = (const float*)d_in[20 + 4 * i];
    sf.box[i] = (const float*)d_in[21 + 4 * i];
    sf.ang[i] = (const float*)d_in[22 + 4 * i];
    sf.ctr[i] = (const float*)d_in[23 + 4 * i];
  }

  char* base = (char*)d_ws;
  size_t off = 0;
  auto alloc = [&](size_t bytes) -> void* {
    void* p = base + off;
    off += (bytes + 255) & ~((size_t)255);
    return p;
  };

  float* sig_cls = (float*)alloc(sizeof(float) * NTOT * NC);
  float* tbox    = (float*)alloc(sizeof(float) * NTOT * 5);
  float* sig_iou = (float*)alloc(sizeof(float) * NTOT);
  float* maxv    = (float*)alloc(sizeof(float) * NTOT);
  float* rowmax  = (float*)alloc(sizeof(float) * NTOT);
  int*   clsind  = (int*)alloc(sizeof(int) * NTOT);
  unsigned char* pos = (unsigned char*)alloc(NTOT);
  unsigned char* hnm = (unsigned char*)alloc(NTOT);
  float* locm = (float*)alloc(sizeof(float) * NTOT);
  float* loct = (float*)alloc(sizeof(float) * NTOT * 5);
  float* locw = (float*)alloc(sizeof(float) * NTOT);
  unsigned long long* keys = (unsigned long long*)alloc(sizeof(unsigned long long) * NTOT);
  float* p_cnt = (float*)alloc(sizeof(float) * NBLK);
  float* p_sum = (float*)alloc(sizeof(float) * NBLK);
  float* p_sq  = (float*)alloc(sizeof(float) * NBLK);
  float* p_pos = (float*)alloc(sizeof(float) * NBLK);
  float* sc    = (float*)alloc(sizeof(float) * 16);
  int*   pidx  = (int*)alloc(sizeof(int) * KPOS);
  int*   pvalid= (int*)alloc(sizeof(int) * KPOS);
  float* pv    = (float*)alloc(sizeof(float) * KPOS);
  float* pbox  = (float*)alloc(sizeof(float) * KPOS * 5);
  int*   pcls  = (int*)alloc(sizeof(int) * KPOS);
  float* pscl  = (float*)alloc(sizeof(float) * KPOS);
  float* Bpad  = (float*)alloc(sizeof(float) * KPOS * 16);
  int*   hidx  = (int*)alloc(sizeof(int) * KHN);
  int*   hvalid= (int*)alloc(sizeof(int) * KHN);
  float* hpts  = (float*)alloc(sizeof(float) * KHN * 2);
  float* hbox  = (float*)alloc(sizeof(float) * KHN * 5);
  int*   hcls  = (int*)alloc(sizeof(int) * KHN);
  float* hscl  = (float*)alloc(sizeof(float) * KHN);
  float* Wm    = (float*)alloc(sizeof(float) * KHN * KPOS);
  int*   potential = (int*)alloc(sizeof(int) * KHN);
  float* aggr  = (float*)alloc(sizeof(float) * KHN * 16);
  float* pc    = (float*)alloc(sizeof(float) * NBLK);
  float* pl    = (float*)alloc(sizeof(float) * NBLK);
  float* pi_   = (float*)alloc(sizeof(float) * NBLK);
  float* lcp   = (float*)alloc(sizeof(float) * NBLK);
  float* lbp   = (float*)alloc(sizeof(float) * NBLK);
  float* lip   = (float*)alloc(sizeof(float) * NBLK);

  k_convert<<<NBLK, 256, 0, stream>>>(tf, sig_cls, tbox, sig_iou, maxv, rowmax, clsind, p_cnt, p_sum, p_sq);
  k_stats<<<1, 512, 0, stream>>>(p_cnt, p_sum, p_sq, sc);
  k_masks<<<NBLK, 256, 0, stream>>>(maxv, sc, pos, hnm, p_pos);
  k_fallback<<<1, 1024, 0, stream>>>(p_pos, sc, maxv, pos, keys);
  k_targets<<<NBLK, 256, 0, stream>>>(pos, tbox, loct, locm);
  k_mkkeys<<<NBLK, 256, 0, stream>>>(maxv, pos, keys);
  k_select<<<1, 1024, 0, stream>>>(keys, KPOS, pidx);
  k_mkkeys<<<NBLK, 256, 0, stream>>>(maxv, hnm, keys);
  k_select<<<1, 1024, 0, stream>>>(keys, KHN, hidx);
  k_gpos<<<1, 256, 0, stream>>>(pidx, pos, maxv, clsind, tbox, pvalid, pv, pbox, pcls, pscl, Bpad);
  k_ghn<<<KHN / 256, 256, 0, stream>>>(hidx, hnm, tbox, clsind, hvalid, hpts, hbox, hcls, hscl);
  k_iou<<<KHN, 256, 0, stream>>>(hbox, hpts, hvalid, hcls, hscl, pbox, pvalid, pv, pcls, pscl, Wm, potential);
  k_wmma<<<KHN / 16, 32, 0, stream>>>(Wm, Bpad, aggr);
  k_aggfin<<<KHN / 256, 256, 0, stream>>>(potential, aggr, hpts, hidx, loct, locm);
  k_fsum<<<NBLK, 256, 0, stream>>>(pos, hnm, rowmax, locm, locw, pc, pl, pi_);
  k_fsfin<<<1, 512, 0, stream>>>(pc, pl, pi_, sc);
  k_student<<<NBLK, 256, 0, stream>>>(sf, pos, hnm, sig_cls, sig_iou, loct, locm, locw, lcp, lbp, lip);
  k_out<<<1, 512, 0, stream>>>(lcp, lbp, lip, sc, (float*)d_out);
}